// SNNPolicy_37632503447808
// MI455X (gfx1250) — compile-verified
//
#include <hip/hip_runtime.h>
#include <hip/hip_bf16.h>
#include <math.h>

typedef __attribute__((ext_vector_type(2))) float v2f;
typedef __attribute__((ext_vector_type(8))) float v8f;

#define DD   8192
#define BB   8
#define HID  256
#define TDIM 128

__device__ __forceinline__ void atomAddF(float* p, float v) {
  __hip_atomic_fetch_add(p, v, __ATOMIC_RELAXED, __HIP_MEMORY_SCOPE_AGENT);
}

// ---------------------------------------------------------------------------
// Kernel 0: SNN collapse coefficients c[k][l] = W1[:,k] . W2[:,l]  (9 dots of 256)
//           + time-embedding MLP  t_out = silu(emb@t_w1+b1)@t_w2+b2   [8,256]
// ---------------------------------------------------------------------------
__global__ void k0_coef_temb(const float* __restrict__ W1, const float* __restrict__ W2,
                             const float* __restrict__ t,
                             const float* __restrict__ t_w1, const float* __restrict__ t_b1,
                             const float* __restrict__ t_w2, const float* __restrict__ t_b2,
                             float* __restrict__ coef, float* __restrict__ t_out) {
  __shared__ float red[256];
  __shared__ float emb[BB][TDIM];
  __shared__ float hid[BB][HID];
  const int tid = threadIdx.x;

  // 9 coefficients (tid = hidden index o)
  for (int kl = 0; kl < 9; ++kl) {
    const int k = kl / 3, l = kl % 3;
    red[tid] = W1[tid * 3 + k] * W2[tid * 3 + l];
    __syncthreads();
    for (int s = 128; s > 0; s >>= 1) {
      if (tid < s) red[tid] += red[tid + s];
      __syncthreads();
    }
    if (tid == 0) coef[kl] = red[0];
    __syncthreads();
  }

  // timestep embedding: emb[b][j] = cos/sin(t_b * exp(-ln(1e4)*i/64))
  if (tid < TDIM) {
    const int j = tid;
    const int i = (j < 64) ? j : (j - 64);
    const float fr = __expf(-logf(10000.0f) * (float)i / 64.0f);
    for (int b = 0; b < BB; ++b) {
      const float a = t[b] * fr;
      emb[b][j] = (j < 64) ? __cosf(a) : __sinf(a);
    }
  }
  __syncthreads();

  // hid = silu(emb @ t_w1 + t_b1)   (tid = hidden column)
  {
    float acc[BB];
    for (int b = 0; b < BB; ++b) acc[b] = t_b1[tid];
    for (int j = 0; j < TDIM; ++j) {
      const float w = t_w1[j * HID + tid];
      for (int b = 0; b < BB; ++b) acc[b] += emb[b][j] * w;
    }
    for (int b = 0; b < BB; ++b) {
      const float v = acc[b];
      hid[b][tid] = v / (1.0f + __expf(-v));
    }
  }
  __syncthreads();

  // t_out = hid @ t_w2 + t_b2
  {
    float acc[BB];
    for (int b = 0; b < BB; ++b) acc[b] = t_b2[tid];
    for (int h = 0; h < HID; ++h) {
      const float w = t_w2[h * HID + tid];
      for (int b = 0; b < BB; ++b) acc[b] += hid[b][h] * w;
    }
    for (int b = 0; b < BB; ++b) t_out[b * HID + tid] = acc[b];
  }
}

// ---------------------------------------------------------------------------
__global__ void k_zero(float* __restrict__ p, int n) {
  const int i = blockIdx.x * blockDim.x + threadIdx.x;
  if (i < n) p[i] = 0.0f;
}

// COO SpMV over all 8 batch rows: out[b][row] += val * in[b][col]
__global__ void k_spmv_atomic(const int* __restrict__ idx, const float* __restrict__ val,
                              const float* __restrict__ xin, float* __restrict__ out, int E) {
  const int e = blockIdx.x * blockDim.x + threadIdx.x;
  if (e >= E) return;
  const int   row = idx[e];
  const int   col = idx[E + e];
  const float v   = val[e];
#pragma unroll
  for (int b = 0; b < BB; ++b)
    atomAddF(&out[b * DD + row], v * xin[b * DD + col]);
}

// y = c00*u + c10*d + c20*p ; s_b = c01*u + c11*d + c21*p ; s_c = c02*u + ...
__global__ void k_combine(const float* __restrict__ x, const float* __restrict__ dbuf,
                          const float* __restrict__ pbuf, const float* __restrict__ c,
                          float* __restrict__ y, float* __restrict__ sb,
                          float* __restrict__ sc, int n) {
  const int i = blockIdx.x * blockDim.x + threadIdx.x;
  if (i >= n) return;
  const float u = x[i], dd = dbuf[i], pp = pbuf[i];
  y[i]  = c[0] * u + c[3] * dd + c[6] * pp;
  sb[i] = c[1] * u + c[4] * dd + c[7] * pp;
  sc[i] = c[2] * u + c[5] * dd + c[8] * pp;
}

// ---------------------------------------------------------------------------
// GEMM h += A[8,8192] @ W[8192,256] with V_WMMA_F32_16X16X4_F32.
// grid = (16 N-tiles, 32 K-chunks); K-chunk of 256 per block, atomic reduce.
// A rows 8..15 are zero padding: load clamped row unconditionally, mask via
// select (no exec divergence in the hot loop).
// ---------------------------------------------------------------------------
__global__ void k_gemm_map(const float* __restrict__ A, const float* __restrict__ W,
                           float* __restrict__ H) {
  const int tn   = blockIdx.x;
  const int kc   = blockIdx.y;
  const int lane = threadIdx.x;       // 0..31
  const int half = lane >> 4;         // K-group select
  const int ml   = lane & 15;         // M row / N col
  const int n    = tn * 16 + ml;
  const bool mvalid = (ml < BB);
  const float* __restrict__ Arow = A + (mvalid ? ml : 0) * DD;  // clamped, in-bounds
  v8f acc = {};
  const int k0 = kc * 256;
#pragma unroll 8
  for (int k = k0; k < k0 + 256; k += 4) {
    const int ka = k + 2 * half;
    const float2 av = *(const float2*)(Arow + ka);   // contiguous b64 load
    v2f a, b;
    a.x = mvalid ? av.x : 0.0f;
    a.y = mvalid ? av.y : 0.0f;
    b.x = W[ka * HID + n];
    b.y = W[(ka + 1) * HID + n];
    acc = __builtin_amdgcn_wmma_f32_16x16x4_f32(false, a, false, b, (short)0, acc,
                                                false, false);
  }
  if (half == 0) {                    // lanes 0..15 carry M rows 0..7
#pragma unroll
    for (int r = 0; r < BB; ++r) atomAddF(&H[r * HID + n], acc[r]);
  }
}

// g = silu((h + map_b + t_out) @ o_w1 + o_b1)   [8,256], one block
__global__ void k_mid(const float* __restrict__ H, const float* __restrict__ map_b,
                      const float* __restrict__ t_out, const float* __restrict__ o_w1,
                      const float* __restrict__ o_b1, float* __restrict__ G) {
  __shared__ float h2[BB][HID];
  const int tid = threadIdx.x;
  for (int b = 0; b < BB; ++b)
    h2[b][tid] = H[b * HID + tid] + map_b[tid] + t_out[b * HID + tid];
  __syncthreads();
  float acc[BB];
  for (int b = 0; b < BB; ++b) acc[b] = o_b1[tid];
  for (int h = 0; h < HID; ++h) {
    const float w = o_w1[h * HID + tid];
    for (int b = 0; b < BB; ++b) acc[b] += h2[b][h] * w;
  }
  for (int b = 0; b < BB; ++b) {
    const float v = acc[b];
    G[b * HID + tid] = v / (1.0f + __expf(-v));
  }
}

// out = G[8,256] @ o_w2[256,8192] + o_b2, WMMA f32, 512 N-tiles
__global__ void k_gemm_out(const float* __restrict__ G, const float* __restrict__ W,
                           const float* __restrict__ bias, float* __restrict__ OUT) {
  const int tn   = blockIdx.x;
  const int lane = threadIdx.x;
  const int half = lane >> 4;
  const int ml   = lane & 15;
  const int n    = tn * 16 + ml;
  const bool mvalid = (ml < BB);
  const float* __restrict__ Grow = G + (mvalid ? ml : 0) * HID;
  v8f acc = {};
#pragma unroll 8
  for (int k = 0; k < HID; k += 4) {
    const int ka = k + 2 * half;
    const float2 av = *(const float2*)(Grow + ka);
    v2f a, b;
    a.x = mvalid ? av.x : 0.0f;
    a.y = mvalid ? av.y : 0.0f;
    b.x = W[ka * DD + n];
    b.y = W[(ka + 1) * DD + n];
    acc = __builtin_amdgcn_wmma_f32_16x16x4_f32(false, a, false, b, (short)0, acc,
                                                false, false);
  }
  if (half == 0) {
    const float bn = bias[n];
#pragma unroll
    for (int r = 0; r < BB; ++r) OUT[r * DD + n] = acc[r] + bn;
  }
}

// ---------------------------------------------------------------------------
extern "C" void kernel_launch(void* const* d_in, const int* in_sizes, int n_in,
                              void* d_out, int out_size, void* d_ws, size_t ws_size,
                              hipStream_t stream) {
  const float* x      = (const float*)d_in[0];
  const float* t      = (const float*)d_in[1];
  const int*   ld_idx = (const int*)  d_in[2];
  const float* ld_val = (const float*)d_in[3];
  const int*   lu_idx = (const int*)  d_in[4];
  const float* lu_val = (const float*)d_in[5];
  const float* W1     = (const float*)d_in[6];
  const float* W2     = (const float*)d_in[7];
  const float* t_w1   = (const float*)d_in[8];
  const float* t_b1   = (const float*)d_in[9];
  const float* t_w2   = (const float*)d_in[10];
  const float* t_b2   = (const float*)d_in[11];
  const float* map_w  = (const float*)d_in[12];
  const float* map_b  = (const float*)d_in[13];
  const float* o_w1   = (const float*)d_in[14];
  const float* o_b1   = (const float*)d_in[15];
  const float* o_w2   = (const float*)d_in[16];
  const float* o_b2   = (const float*)d_in[17];
  float* out = (float*)d_out;
  const int E = in_sizes[3];          // 131072

  float* ws    = (float*)d_ws;
  float* t_out = ws;                  // 2048
  float* coef  = ws + 2048;           // 16 (9 used)
  float* dbuf  = ws + 4096;           // 65536  (Ld x)
  float* pbuf  = dbuf + BB * DD;      // 65536  (Lu x)
  float* sb    = pbuf + BB * DD;      // 65536
  float* sc    = sb   + BB * DD;      // 65536
  float* ybuf  = sc   + BB * DD;      // 65536  (x_out)
  float* hbuf  = ybuf + BB * DD;      // 2048
  float* gbuf  = hbuf + BB * HID;     // 2048

  k0_coef_temb<<<1, 256, 0, stream>>>(W1, W2, t, t_w1, t_b1, t_w2, t_b2, coef, t_out);

  k_zero<<<(2 * BB * DD + 255) / 256, 256, 0, stream>>>(dbuf, 2 * BB * DD);
  k_zero<<<(BB * HID + 255) / 256, 256, 0, stream>>>(hbuf, BB * HID);

  const int spmv_blocks = (E + 255) / 256;
  k_spmv_atomic<<<spmv_blocks, 256, 0, stream>>>(ld_idx, ld_val, x, dbuf, E);
  k_spmv_atomic<<<spmv_blocks, 256, 0, stream>>>(lu_idx, lu_val, x, pbuf, E);

  k_combine<<<(BB * DD + 255) / 256, 256, 0, stream>>>(x, dbuf, pbuf, coef,
                                                       ybuf, sb, sc, BB * DD);

  k_spmv_atomic<<<spmv_blocks, 256, 0, stream>>>(ld_idx, ld_val, sb, ybuf, E);
  k_spmv_atomic<<<spmv_blocks, 256, 0, stream>>>(lu_idx, lu_val, sc, ybuf, E);

  dim3 gmap(HID / 16, 32);            // 16 N-tiles x 32 K-chunks
  k_gemm_map<<<gmap, 32, 0, stream>>>(ybuf, map_w, hbuf);

  k_mid<<<1, 256, 0, stream>>>(hbuf, map_b, t_out, o_w1, o_b1, gbuf);

  k_gemm_out<<<DD / 16, 32, 0, stream>>>(gbuf, o_w2, o_b2, out);
}